// AF_LSTM_85023172592231
// MI455X (gfx1250) — compile-verified
//
#include <hip/hip_runtime.h>
#include <math.h>

// ---------------------------------------------------------------------------
// AF-LSTM forward for MI455X (gfx1250, wave32)
//   - fp32 matrix path: v_wmma_f32_16x16x4_f32
//   - Tensor Data Mover stages shared A-panels into LDS (tensor_load_to_lds,
//     s_wait_tensorcnt), with TDM padding to kill LDS bank conflicts
// ---------------------------------------------------------------------------

typedef __attribute__((ext_vector_type(2))) float v2f;
typedef __attribute__((ext_vector_type(8))) float v8f;
typedef __attribute__((ext_vector_type(4))) unsigned int v4u;
typedef __attribute__((ext_vector_type(8))) int v8i;
typedef __attribute__((ext_vector_type(4))) int v4i;

#define Bsz 64
#define Tsz 256
#define Dsz 512
#define Hsz 512
#define Asz 4
#define G4H 2048     // 4*H
#define KDIM 512     // every GEMM in this net has K = 512

// Effective LDS row pitch after TDM padding: 256 data + 2 pad + 256 data + 2 pad
#define APITCH 516

// ---------------------------------------------------------------------------
// TDM: DMA a 16-row x 512-col fp32 panel (row stride = lda elements) from
// global into LDS, inserting a 2-DWORD pad every 256 DWORDs (bank staggering).
// D# layout per CDNA5 ISA ch.8 (group0: count/lds/global/type, group1: sizes).
// This toolchain (clang-23 lane) uses the 6-arg builtin form.
// ---------------------------------------------------------------------------
__device__ __forceinline__ void tdm_load_panel16(const float* gsrc,
                                                 unsigned lda_elems,
                                                 float* ldsDst)
{
    unsigned long long ga = (unsigned long long)(uintptr_t)gsrc;
    unsigned la = (unsigned)(uintptr_t)ldsDst;   // flat LDS addr: low 32 bits = LDS offset
    v4u g0; v8i g1; v4i g2; v4i g3; v8i g4;
    g0[0] = 1u;                                              // count=1, user desc
    g0[1] = la;                                              // lds_addr (bytes)
    g0[2] = (unsigned)ga;                                    // global_addr[31:0]
    g0[3] = (unsigned)((ga >> 32) & 0x01FFFFFFull) | (2u << 30); // addr[56:32] | type=2
    g1[0] = (int)((2u << 16)          // data_size = 4 bytes
                | (1u << 20)          // pad_enable
                | (7u << 22)          // pad_interval = 256 DWORDs
                | (1u << 25));        // pad_amount   = 2 DWORDs
    g1[1] = (int)((unsigned)(KDIM & 0xFFFF) << 16);          // tensor_dim0 lo16 = 512
    g1[2] = (int)(16u << 16);                                // td0 hi16=0 | tensor_dim1 lo16=16
    g1[3] = (int)((unsigned)(KDIM & 0xFFFF) << 16);          // td1 hi16=0 | tile_dim0 = 512
    g1[4] = 16;                                              // tile_dim1 = 16 | tile_dim2 = 0
    g1[5] = (int)lda_elems;                                  // tensor_dim0_stride lo32
    g1[6] = 0;                                               // stride hi | td1_stride lo
    g1[7] = 0;
    g2[0] = g2[1] = g2[2] = g2[3] = 0;
    g3[0] = g3[1] = g3[2] = g3[3] = 0;
    g4[0] = g4[1] = g4[2] = g4[3] = 0;
    g4[4] = g4[5] = g4[6] = g4[7] = 0;
    __builtin_amdgcn_tensor_load_to_lds(g0, g1, g2, g3, g4, 0);
}

// ---------------------------------------------------------------------------
// Generic NT GEMM (K fixed = 512):
//   C[M,N] = act( A[M,K] @ Bt[N,K]^T + bias[N] + Dadd[M,N] )
// Block = 256 threads = 8 waves; block computes a 16(M) x 128(N) macro-tile.
// The shared 16xK A-panel is TDM-staged into LDS once per block; each wave
// runs one 16x16 tile on v_wmma_f32_16x16x4_f32. grid = Mtiles * nblkN.
// ---------------------------------------------------------------------------
__global__ void gemm_nt_kernel(float* __restrict__ C, int ldc,
                               const float* __restrict__ A, int lda,
                               const float* __restrict__ Bt, int ldb,
                               const float* __restrict__ bias,
                               const float* __restrict__ Dadd, int ldd,
                               int nblkN, int act)
{
    __shared__ float ldsA[16 * APITCH];
    const int wv   = threadIdx.x >> 5;
    const int lane = threadIdx.x & 31;
    const int hl   = lane >> 4;          // K-pair select: {0,1} vs {2,3}
    const int l    = lane & 15;
    const int tm   = blockIdx.x / nblkN;             // M tile (16 rows)
    const int tn   = (blockIdx.x % nblkN) * 8 + wv;  // N tile per wave

    if (threadIdx.x < 32) {
        if (threadIdx.x == 0)                      // touch a TDM pad slot so the
            ((volatile float*)ldsA)[514] = 0.0f;   // store-less LDS isn't folded
        tdm_load_panel16(A + (size_t)tm * 16 * lda, (unsigned)lda, ldsA);
        __builtin_amdgcn_s_wait_tensorcnt((short)0);
    }
    __syncthreads();

    const float* __restrict__ brow = Bt + (size_t)(tn * 16 + l) * ldb + 2 * hl;
    __builtin_prefetch(brow, 0, 1);
    const float* aL = ldsA + l * APITCH + 2 * hl;

    v8f acc = {};
#pragma unroll 4
    for (int k = 0; k < 256; k += 4) {             // first half: no pad offset
        v2f a, b;
        a.x = aL[k];       a.y = aL[k + 1];
        b.x = brow[k];     b.y = brow[k + 1];
        acc = __builtin_amdgcn_wmma_f32_16x16x4_f32(
            false, a, false, b, (short)0, acc, false, false);
    }
#pragma unroll 4
    for (int k = 256; k < 512; k += 4) {           // second half: +2 DWORD pad
        v2f a, b;
        a.x = aL[k + 2];   a.y = aL[k + 3];
        b.x = brow[k];     b.y = brow[k + 1];
        acc = __builtin_amdgcn_wmma_f32_16x16x4_f32(
            false, a, false, b, (short)0, acc, false, false);
    }

    const int col     = tn * 16 + l;
    const int rowbase = tm * 16 + hl * 8;
#pragma unroll
    for (int i = 0; i < 8; ++i) {
        const int row = rowbase + i;
        float v = acc[i];
        if (bias) v += bias[col];
        if (Dadd) v += Dadd[(size_t)row * ldd + col];
        if (act == 1) v = tanhf(v);
        C[(size_t)row * ldc + col] = v;
    }
}

// ---------------------------------------------------------------------------
// Circular correlation as per-batch circulant matmul on the matrix pipe:
//   m[b,t,n] = sum_k Hs[b,t,k] * s_norm[b, (k - n) & 511]
// Block = (b, tm, 4 N-groups of 8 waves): Hs panel TDM-staged, s_norm row in
// LDS (rotating gather hits consecutive banks across lanes).
// grid = 64 * 16 * 4 = 4096 blocks.
// ---------------------------------------------------------------------------
__global__ void circcorr_kernel(float* __restrict__ Mout,
                                const float* __restrict__ Hs,
                                const float* __restrict__ sn)
{
    __shared__ float ldsA[16 * APITCH];
    __shared__ float sbuf[Dsz];
    const int wv   = threadIdx.x >> 5;
    const int lane = threadIdx.x & 31;
    const int hl   = lane >> 4;
    const int l    = lane & 15;
    const int b    = blockIdx.x >> 6;
    const int rem  = blockIdx.x & 63;
    const int tm   = rem >> 2;                 // T tile (16 rows)
    const int tn   = (rem & 3) * 8 + wv;       // D tile per wave

    sbuf[threadIdx.x]       = sn[b * Dsz + threadIdx.x];
    sbuf[threadIdx.x + 256] = sn[b * Dsz + threadIdx.x + 256];

    if (threadIdx.x < 32) {
        if (threadIdx.x == 0)
            ((volatile float*)ldsA)[514] = 0.0f;
        tdm_load_panel16(Hs + (size_t)b * (Tsz * Dsz) + (size_t)tm * 16 * Dsz,
                         Dsz, ldsA);
        __builtin_amdgcn_s_wait_tensorcnt((short)0);
    }
    __syncthreads();

    const int n = tn * 16 + l;
    const float* aL = ldsA + l * APITCH + 2 * hl;

    v8f acc = {};
#pragma unroll 4
    for (int k = 0; k < 256; k += 4) {
        v2f a, bb;
        a.x = aL[k];      a.y = aL[k + 1];
        const int kk = k + 2 * hl;
        bb.x = sbuf[(kk     - n) & (Dsz - 1)];
        bb.y = sbuf[(kk + 1 - n) & (Dsz - 1)];
        acc = __builtin_amdgcn_wmma_f32_16x16x4_f32(
            false, a, false, bb, (short)0, acc, false, false);
    }
#pragma unroll 4
    for (int k = 256; k < 512; k += 4) {
        v2f a, bb;
        a.x = aL[k + 2];  a.y = aL[k + 3];
        const int kk = k + 2 * hl;
        bb.x = sbuf[(kk     - n) & (Dsz - 1)];
        bb.y = sbuf[(kk + 1 - n) & (Dsz - 1)];
        acc = __builtin_amdgcn_wmma_f32_16x16x4_f32(
            false, a, false, bb, (short)0, acc, false, false);
    }

    float* __restrict__ Cb = Mout + (size_t)b * (Tsz * Dsz);
    const int rowbase = tm * 16 + hl * 8;
#pragma unroll
    for (int i = 0; i < 8; ++i)
        Cb[(size_t)(rowbase + i) * Dsz + tn * 16 + l] = acc[i];
}

// ---------------------------------------------------------------------------
// Elementwise / small kernels
// ---------------------------------------------------------------------------
__global__ void zero_kernel(float* __restrict__ p, int n) {
    int i = blockIdx.x * blockDim.x + threadIdx.x;
    if (i < n) p[i] = 0.0f;
}

__global__ void gather_kernel(float* __restrict__ out,
                              const float* __restrict__ embed,
                              const int* __restrict__ idx)
{
    const int token = blockIdx.x;
    const float4* __restrict__ src =
        (const float4*)(embed + (size_t)idx[token] * Dsz);
    float4* __restrict__ dst = (float4*)(out + (size_t)token * Dsz);
    dst[threadIdx.x] = src[threadIdx.x];
}

__global__ void aspect_stats_kernel(float* __restrict__ mu, float* __restrict__ var,
                                    const float* __restrict__ embed,
                                    const int* __restrict__ s)
{
    const int d = threadIdx.x;
    float sum = 0.0f, sq = 0.0f;
    for (int j = 0; j < Bsz * Asz; ++j) {
        float v = embed[(size_t)s[j] * Dsz + d];
        sum += v; sq += v * v;
    }
    const float m = sum * (1.0f / (Bsz * Asz));
    mu[d]  = m;
    var[d] = sq * (1.0f / (Bsz * Asz)) - m * m;
}

__global__ void aspect_norm_kernel(float* __restrict__ sn,
                                   const float* __restrict__ embed,
                                   const int* __restrict__ s,
                                   const float* __restrict__ mu,
                                   const float* __restrict__ var)
{
    const int b = blockIdx.x, d = threadIdx.x;
    float sum = 0.0f;
    for (int a = 0; a < Asz; ++a)
        sum += embed[(size_t)s[b * Asz + a] * Dsz + d];
    sn[b * Dsz + d] = (sum - (float)Asz * mu[d]) * rsqrtf(var[d] + 1e-5f);
}

__global__ void lstm_gates_kernel(float* __restrict__ Hs, float* __restrict__ c,
                                  const float* __restrict__ z, int t)
{
    const int idx = blockIdx.x * blockDim.x + threadIdx.x;   // b*512 + d
    const int b = idx >> 9, d = idx & 511;
    const float* __restrict__ zb = z + b * G4H;
    const float zi = zb[d], zf = zb[512 + d], zg = zb[1024 + d], zo = zb[1536 + d];
    const float si = 1.0f / (1.0f + expf(-zi));
    const float sf = 1.0f / (1.0f + expf(-zf));
    const float so = 1.0f / (1.0f + expf(-zo));
    const float cn = sf * c[idx] + si * tanhf(zg);
    c[idx] = cn;
    Hs[(size_t)b * (Tsz * Hsz) + (size_t)t * Hsz + d] = so * tanhf(cn);
}

__global__ void attn_time_softmax_kernel(float* __restrict__ r,
                                         const float* __restrict__ S,
                                         const float* __restrict__ Hs)
{
    const int b = blockIdx.x, d = threadIdx.x;
    const float* __restrict__ Sb = S  + (size_t)b * (Tsz * Dsz) + d;
    const float* __restrict__ Hb = Hs + (size_t)b * (Tsz * Dsz) + d;
    float mx = -1e30f;
    for (int t = 0; t < Tsz; ++t) mx = fmaxf(mx, Sb[t * Dsz]);
    float sum = 0.0f;
    for (int t = 0; t < Tsz; ++t) sum += expf(Sb[t * Dsz] - mx);
    const float inv = 1.0f / sum;
    float acc = 0.0f;
    for (int t = 0; t < Tsz; ++t)
        acc += expf(Sb[t * Dsz] - mx) * inv * Hb[t * Dsz];
    r[b * Dsz + d] = acc;
}

__global__ void row_softmax_kernel(float* __restrict__ out,
                                   const float* __restrict__ logits)
{
    __shared__ float red[Dsz];
    const int b = blockIdx.x, d = threadIdx.x;
    const float v = logits[b * Dsz + d];
    red[d] = v; __syncthreads();
    for (int s = 256; s > 0; s >>= 1) {
        if (d < s) red[d] = fmaxf(red[d], red[d + s]);
        __syncthreads();
    }
    const float mx = red[0]; __syncthreads();
    const float e = expf(v - mx);
    red[d] = e; __syncthreads();
    for (int s = 256; s > 0; s >>= 1) {
        if (d < s) red[d] += red[d + s];
        __syncthreads();
    }
    out[b * Dsz + d] = e / red[0];
}

// ---------------------------------------------------------------------------
extern "C" void kernel_launch(void* const* d_in, const int* in_sizes, int n_in,
                              void* d_out, int out_size, void* d_ws, size_t ws_size,
                              hipStream_t stream)
{
    const int*   x      = (const int*)  d_in[0];
    const int*   s      = (const int*)  d_in[1];
    const float* embed  = (const float*)d_in[2];
    const float* W_ih   = (const float*)d_in[3];
    const float* W_hh   = (const float*)d_in[4];
    const float* b_lstm = (const float*)d_in[5];
    const float* w_y    = (const float*)d_in[6];
    const float* w_t    = (const float*)d_in[7];
    const float* w_p    = (const float*)d_in[8];
    const float* w_x    = (const float*)d_in[9];
    const float* w_f    = (const float*)d_in[10];
    const float* b_f    = (const float*)d_in[11];
    float* out = (float*)d_out;

    // ---- workspace layout (floats) ----
    float* ws = (float*)d_ws;
    size_t off = 0;
    float* emb_x = ws + off; off += (size_t)Bsz * Tsz * Dsz;   // reused as m
    float* Zx    = ws + off; off += (size_t)Bsz * Tsz * G4H;   // reused as Y,S
    float* Hs    = ws + off; off += (size_t)Bsz * Tsz * Hsz;
    float* zstep = ws + off; off += (size_t)Bsz * G4H;
    float* cbuf  = ws + off; off += (size_t)Bsz * Hsz;         // zeroed below
    float* zeroh = ws + off; off += 1024;                      // zeroed below
    float* mu    = ws + off; off += Dsz;
    float* var   = ws + off; off += Dsz;
    float* sn    = ws + off; off += (size_t)Bsz * Dsz;
    float* rbuf  = ws + off; off += (size_t)Bsz * Dsz;
    float* tmp1  = ws + off; off += (size_t)Bsz * Dsz;
    float* hstar = ws + off; off += (size_t)Bsz * Dsz;
    float* lgt   = ws + off; off += (size_t)Bsz * Dsz;

    float* m = emb_x;                               // alias after Zx built
    float* Y = Zx;                                  // alias after LSTM done
    float* S = Zx + (size_t)Bsz * Tsz * Dsz;

    // deterministic init of persistent state (cbuf + zeroh contiguous)
    zero_kernel<<<(Bsz * Hsz + 1024 + 255) / 256, 256, 0, stream>>>(
        cbuf, Bsz * Hsz + 1024);

    // 1) embedding gather for the sentence
    gather_kernel<<<Bsz * Tsz, 128, 0, stream>>>(emb_x, embed, x);

    // 2) aspect batch-norm + sum over aspect tokens
    aspect_stats_kernel<<<1, Dsz, 0, stream>>>(mu, var, embed, s);
    aspect_norm_kernel<<<Bsz, Dsz, 0, stream>>>(sn, embed, s, mu, var);

    // 3) hoisted input projection: Zx = emb_x @ W_ih^T + b  [16384 x 2048]
    gemm_nt_kernel<<<(Bsz * Tsz / 16) * (G4H / 16 / 8), 256, 0, stream>>>(
        Zx, G4H, emb_x, Dsz, W_ih, Dsz, b_lstm, nullptr, 0,
        G4H / 16 / 8, 0);

    // 4) sequential LSTM: per step only the recurrent GEMM (64x2048x512) is
    //    on the critical path; W_hh stays L2-resident across all 256 steps
    for (int t = 0; t < Tsz; ++t) {
        const float* hprev = (t == 0) ? zeroh : (Hs + (size_t)(t - 1) * Hsz);
        const int lda = (t == 0) ? 0 : (Tsz * Hsz);
        gemm_nt_kernel<<<(Bsz / 16) * (G4H / 16 / 8), 256, 0, stream>>>(
            zstep, G4H, hprev, lda, W_hh, Hsz, nullptr,
            Zx + (size_t)t * G4H, Tsz * G4H,
            G4H / 16 / 8, 0);
        lstm_gates_kernel<<<Bsz * Hsz / 256, 256, 0, stream>>>(Hs, cbuf, zstep, t);
    }

    // 5) FFT circular correlation as direct circulant matmul on matrix pipe
    circcorr_kernel<<<Bsz * (Tsz / 16) * (Dsz / 16 / 8), 256, 0, stream>>>(
        m, Hs, sn);

    // 6) attention chain
    gemm_nt_kernel<<<(Bsz * Tsz / 16) * (Dsz / 16 / 8), 256, 0, stream>>>(
        Y, Dsz, m, Dsz, w_y, Dsz, nullptr, nullptr, 0, Dsz / 16 / 8, 1);
    gemm_nt_kernel<<<(Bsz * Tsz / 16) * (Dsz / 16 / 8), 256, 0, stream>>>(
        S, Dsz, Y, Dsz, w_t, Dsz, nullptr, nullptr, 0, Dsz / 16 / 8, 0);
    attn_time_softmax_kernel<<<Bsz, Dsz, 0, stream>>>(rbuf, S, Hs);

    gemm_nt_kernel<<<(Bsz / 16) * (Dsz / 16 / 8), 256, 0, stream>>>(
        tmp1, Dsz, rbuf, Dsz, w_p, Dsz, nullptr, nullptr, 0, Dsz / 16 / 8, 0);
    gemm_nt_kernel<<<(Bsz / 16) * (Dsz / 16 / 8), 256, 0, stream>>>(
        hstar, Dsz, Hs + (size_t)(Tsz - 1) * Hsz, Tsz * Hsz, w_x, Dsz,
        nullptr, tmp1, Dsz, Dsz / 16 / 8, 1);

    // 7) output: softmax(h* @ w_f^T + b_f)
    gemm_nt_kernel<<<(Bsz / 16) * (Dsz / 16 / 8), 256, 0, stream>>>(
        lgt, Dsz, hstar, Dsz, w_f, Dsz, b_f, nullptr, 0, Dsz / 16 / 8, 0);
    row_softmax_kernel<<<Bsz, Dsz, 0, stream>>>(out, lgt);
}